// OICROutputLayers_790273982473
// MI455X (gfx1250) — compile-verified
//
#include <hip/hip_runtime.h>
#include <hip/hip_bf16.h>
#include <math.h>

#define R 2048
#define KCLS 20
#define SSTRIDE 21          // scores row stride (K+1)
#define IMGW 1333.0f
#define IMGH 800.0f
#define NMS_T 0.5f
#define SCORE_T 0.5f
#define NTOP 100
#define NEG_INF (-__builtin_inff())

typedef __attribute__((ext_vector_type(2))) float v2f;
typedef __attribute__((ext_vector_type(8))) float v8f;

__device__ __forceinline__ float clampf(float v, float lo, float hi) {
    return fminf(fmaxf(v, lo), hi);
}

// final ordering: score descending, flat-index ascending (matches lax.top_k ties)
__device__ __forceinline__ bool before_(float ka, int va, float kb, int vb) {
    return (ka > kb) || (ka == kb && va < vb);
}

// ---------------------------------------------------------------------------
// Kernel A: clip boxes, precompute w/h/area, finiteness mask
// ---------------------------------------------------------------------------
__global__ void nms_preprocess(const float* __restrict__ boxes,
                               const float* __restrict__ scores,
                               float* __restrict__ bx,
                               float* __restrict__ aw, float* __restrict__ ah,
                               float* __restrict__ aarea,
                               unsigned* __restrict__ finite) {
    int r = blockIdx.x * blockDim.x + threadIdx.x;
    if (r >= R) return;
    float x1 = boxes[r * 4 + 0], y1 = boxes[r * 4 + 1];
    float x2 = boxes[r * 4 + 2], y2 = boxes[r * 4 + 3];
    bool fin = isfinite(x1) && isfinite(y1) && isfinite(x2) && isfinite(y2);
#pragma unroll 3
    for (int c = 0; c < SSTRIDE; ++c) fin = fin && isfinite(scores[r * SSTRIDE + c]);
    float cx1 = clampf(x1, 0.f, IMGW), cy1 = clampf(y1, 0.f, IMGH);
    float cx2 = clampf(x2, 0.f, IMGW), cy2 = clampf(y2, 0.f, IMGH);
    bx[r * 4 + 0] = cx1; bx[r * 4 + 1] = cy1; bx[r * 4 + 2] = cx2; bx[r * 4 + 3] = cy2;
    float w = cx2 - cx1, h = cy2 - cy1;
    aw[r] = w; ah[r] = h; aarea[r] = w * h;
    finite[r] = fin ? 1u : 0u;
}

// ---------------------------------------------------------------------------
// Kernel B: per-class bitonic sort (desc score, asc index), gather sorted SoA
// ---------------------------------------------------------------------------
__global__ void nms_sort_class(const float* __restrict__ scores,
                               const float* __restrict__ bx,
                               const float* __restrict__ aw, const float* __restrict__ ah,
                               const float* __restrict__ aarea,
                               const unsigned* __restrict__ finite,
                               float* __restrict__ sx1, float* __restrict__ sy1,
                               float* __restrict__ sx2, float* __restrict__ sy2,
                               float* __restrict__ sw, float* __restrict__ sh,
                               float* __restrict__ sarea, float* __restrict__ sscore,
                               unsigned* __restrict__ sidx, unsigned* __restrict__ svalid) {
    int c = blockIdx.x;
    __shared__ float key[R];
    __shared__ int   val[R];
    for (int p = threadIdx.x; p < R; p += blockDim.x) {
        key[p] = scores[p * SSTRIDE + c];
        val[p] = p;
    }
    for (int k = 2; k <= R; k <<= 1) {
        for (int j = k >> 1; j > 0; j >>= 1) {
            __syncthreads();
            int t = threadIdx.x;               // 1024 threads = R/2 pairs
            int i = ((t & ~(j - 1)) << 1) | (t & (j - 1));
            int p = i | j;
            bool desc = (i & k) == 0;          // final pass (k==R) => descending
            float ki = key[i], kp = key[p];
            int   vi = val[i], vp = val[p];
            bool sw2 = desc ? before_(kp, vp, ki, vi) : before_(ki, vi, kp, vp);
            if (sw2) { key[i] = kp; key[p] = ki; val[i] = vp; val[p] = vi; }
        }
    }
    __syncthreads();
    for (int p = threadIdx.x; p < R; p += blockDim.x) {
        int o = val[p];
        float s = key[p];
        int g = c * R + p;
        sx1[g] = bx[o * 4 + 0]; sy1[g] = bx[o * 4 + 1];
        sx2[g] = bx[o * 4 + 2]; sy2[g] = bx[o * 4 + 3];
        sw[g] = aw[o]; sh[g] = ah[o]; sarea[g] = aarea[o];
        sscore[g] = s; sidx[g] = (unsigned)o;
        svalid[g] = ((s > SCORE_T) && finite[o]) ? 1u : 0u;
    }
}

// ---------------------------------------------------------------------------
// Kernel C: pairwise IoU>thresh bitmask via WMMA rank-2 tiles.
// Block = (class, 128-row slab): 8 waves own one 16-row tile each.
// The whole class column SoA (7 x 2048 f32 = 56KB) is staged into LDS once
// per block with GLOBAL_LOAD_ASYNC_TO_LDS_B128 (ASYNCcnt DMA path), giving
// 8x reuse of the staged data. Per 16x16 tile: 5x v_wmma_f32_16x16x4_f32
// compute cross-difference matrices x2_i-x1_j, x2_j-x1_i, y2_i-y1_j,
// y2_j-y1_i and area_i+area_j (rank-2 outer products; exact in f32).
// ---------------------------------------------------------------------------
__global__ void __launch_bounds__(256)
nms_iou_mask(const float* __restrict__ sx1, const float* __restrict__ sy1,
             const float* __restrict__ sx2, const float* __restrict__ sy2,
             const float* __restrict__ sw, const float* __restrict__ sh,
             const float* __restrict__ sarea,
             unsigned* __restrict__ mask) {
    int c    = blockIdx.x >> 4;        // /16
    int rblk = blockIdx.x & 15;
    int lane = threadIdx.x & 31;
    int wave = threadIdx.x >> 5;
    int cbase = c * R;

    __shared__ float smem[7 * R];      // 57344 B, well under 320KB/WGP
    unsigned smemBase = (unsigned)(uintptr_t)smem;     // LDS byte address

    // --- async DMA stage: 7 arrays x 8KB, 2x B128 per thread per array ---
    const float* srcs[7] = { sx1, sy1, sx2, sy2, sw, sh, sarea };
#pragma unroll
    for (int a = 0; a < 7; ++a) {
        unsigned long long gbase = (unsigned long long)(uintptr_t)(srcs[a] + cbase);
        unsigned off0 = threadIdx.x * 16u;             // 256 threads * 16B = 4KB
        unsigned lds0 = smemBase + (unsigned)a * 8192u + off0;
        unsigned long long ga0 = gbase + off0;
        asm volatile("global_load_async_to_lds_b128 %0, %1, off"
                     :: "v"(lds0), "v"(ga0) : "memory");
        unsigned lds1 = lds0 + 4096u;
        unsigned long long ga1 = ga0 + 4096u;
        asm volatile("global_load_async_to_lds_b128 %0, %1, off"
                     :: "v"(lds1), "v"(ga1) : "memory");
    }
    asm volatile("s_wait_asynccnt 0x0" ::: "memory");
    __syncthreads();

    float* LX1 = smem;            float* LY1 = smem + R;
    float* LX2 = smem + 2 * R;    float* LY2 = smem + 3 * R;
    float* LW  = smem + 4 * R;    float* LH  = smem + 5 * R;
    float* LA  = smem + 6 * R;

    int rtg = rblk * 8 + wave;         // this wave's 16-row tile: 0..127
    int r0  = rtg * 16;
    int l15 = lane & 15;
    bool hi = lane >= 16;
    float z = 0.f, one = hi ? 0.f : 1.f;   // zero K=2,3 contribution on hi lanes
    // Row-side operands (A matrices): lanes 0-15 supply K0,K1 for M=lane.
    float a_x2 = hi ? z : LX2[r0 + l15];
    float a_x1 = hi ? z : LX1[r0 + l15];
    float a_y2 = hi ? z : LY2[r0 + l15];
    float a_y1 = hi ? z : LY1[r0 + l15];
    float a_ar = hi ? z : LA[r0 + l15];
    // Per-row clamp operands hoisted into registers (row rl = v + 8*hi).
    float rwv[8], rhv[8];
#pragma unroll
    for (int v = 0; v < 8; ++v) {
        int rl = v + (hi ? 8 : 0);
        rwv[v] = LW[r0 + rl];
        rhv[v] = LH[r0 + rl];
    }

    for (int ct = 0; ct < 128; ++ct) {
        int ci_ = ct * 16 + l15;
        float cx1 = LX1[ci_], cy1 = LY1[ci_], cx2 = LX2[ci_], cy2 = LY2[ci_];
        float cw = LW[ci_], ch = LH[ci_], ca = LA[ci_];

        v2f aDX1 = { a_x2, one };           v2f bDX1 = { one, hi ? z : -cx1 };
        v2f aDX2 = { one, hi ? z : -a_x1 }; v2f bDX2 = { hi ? z : cx2, one };
        v2f aDY1 = { a_y2, one };           v2f bDY1 = { one, hi ? z : -cy1 };
        v2f aDY2 = { one, hi ? z : -a_y1 }; v2f bDY2 = { hi ? z : cy2, one };
        v2f aS   = { a_ar, one };           v2f bS   = { one, hi ? z : ca };

        v8f zc = {};
        v8f dx1 = __builtin_amdgcn_wmma_f32_16x16x4_f32(false, aDX1, false, bDX1, (short)0, zc, false, false);
        v8f dx2 = __builtin_amdgcn_wmma_f32_16x16x4_f32(false, aDX2, false, bDX2, (short)0, zc, false, false);
        v8f dy1 = __builtin_amdgcn_wmma_f32_16x16x4_f32(false, aDY1, false, bDY1, (short)0, zc, false, false);
        v8f dy2 = __builtin_amdgcn_wmma_f32_16x16x4_f32(false, aDY2, false, bDY2, (short)0, zc, false, false);
        v8f sm  = __builtin_amdgcn_wmma_f32_16x16x4_f32(false, aS,   false, bS,   (short)0, zc, false, false);

#pragma unroll
        for (int v = 0; v < 8; ++v) {
            float wv = fminf(fminf(rwv[v], cw), fminf(dx1[v], dx2[v]));
            float hv = fminf(fminf(rhv[v], ch), fminf(dy1[v], dy2[v]));
            float inter = fmaxf(wv, 0.f) * fmaxf(hv, 0.f);
            float uni = sm[v] - inter;
            bool pred = inter > NMS_T * fmaxf(uni, 1e-9f);
            unsigned bal = (unsigned)__ballot(pred);   // wave32: [15:0]=row v, [31:16]=row v+8
            if (lane == v) {
                unsigned shft = (unsigned)(ct & 1) * 16u;
                unsigned lo16 = (bal & 0xffffu) << shft;
                unsigned hi16 = (bal >> 16) << shft;
                unsigned wword = (unsigned)(ct >> 1);
                if (lo16) atomicOr(&mask[((size_t)(cbase + r0 + v)) * 64 + wword], lo16);
                if (hi16) atomicOr(&mask[((size_t)(cbase + r0 + v + 8)) * 64 + wword], hi16);
            }
        }
    }
}

// ---------------------------------------------------------------------------
// Kernel D: greedy sequential scan per class. One wave32 per class; removal
// bitset (2048 bits) lives in registers (2 words per lane); decisions
// broadcast with __shfl. Next mask row prefetched speculatively.
// ---------------------------------------------------------------------------
__global__ void nms_scan(const unsigned* __restrict__ mask,
                         const unsigned* __restrict__ svalid,
                         const float* __restrict__ sscore,
                         const unsigned* __restrict__ sidx,
                         float* __restrict__ masked) {
    int c = blockIdx.x;
    int lane = threadIdx.x;                // 32 threads
    const unsigned* sval = svalid + (size_t)c * R + lane * 64;
    unsigned vm0 = 0, vm1 = 0;
#pragma unroll 8
    for (int b = 0; b < 32; ++b) {
        vm0 |= (sval[b] & 1u) << b;
        vm1 |= (sval[32 + b] & 1u) << b;
    }
    unsigned rm0 = 0, rm1 = 0, kp0 = 0, kp1 = 0;
    for (int i = 0; i < R; ++i) {
        if (i + 1 < R)
            __builtin_prefetch(mask + ((size_t)(c * R + i + 1)) * 64 + 2 * lane, 0, 0);
        int owner = i >> 6;
        int sel = (i >> 5) & 1;
        int bit = i & 31;
        unsigned rmw = sel ? rm1 : rm0;
        unsigned vmw = sel ? vm1 : vm0;
        int my = ((((rmw >> bit) & 1u) == 0u) && ((vmw >> bit) & 1u)) ? 1 : 0;
        int dec = __shfl(my, owner, 32);
        if (dec) {
            const unsigned* mrow = mask + ((size_t)(c * R + i)) * 64;
            rm0 |= mrow[2 * lane];
            rm1 |= mrow[2 * lane + 1];
            if (lane == owner) { if (sel) kp1 |= 1u << bit; else kp0 |= 1u << bit; }
        }
    }
    const float* ss = sscore + (size_t)c * R;
    const unsigned* si = sidx + (size_t)c * R;
    float* mo = masked + (size_t)c * R;
#pragma unroll 4
    for (int b = 0; b < 64; ++b) {
        int p = lane * 64 + b;
        unsigned kpw = (b < 32) ? kp0 : kp1;
        bool kept = (kpw >> (b & 31)) & 1u;
        mo[si[p]] = kept ? ss[p] : NEG_INF;
    }
}

// ---------------------------------------------------------------------------
// Kernel E1: block-local bitonic sort of 1024 masked scores -> top 100
// ---------------------------------------------------------------------------
__global__ void topk_stage1(const float* __restrict__ masked,
                            float* __restrict__ cs, int* __restrict__ ci) {
    __shared__ float key[1024];
    __shared__ int   val[1024];
    int t = threadIdx.x;
    int base = blockIdx.x * 1024;
    key[t] = masked[base + t];
    val[t] = base + t;                      // flat index = class*R + idx
    for (int k = 2; k <= 1024; k <<= 1) {
        for (int j = k >> 1; j > 0; j >>= 1) {
            __syncthreads();
            if (t < 512) {
                int i = ((t & ~(j - 1)) << 1) | (t & (j - 1));
                int p = i | j;
                bool desc = (i & k) == 0;
                float ki = key[i], kp = key[p];
                int   vi = val[i], vp = val[p];
                bool sw2 = desc ? before_(kp, vp, ki, vi) : before_(ki, vi, kp, vp);
                if (sw2) { key[i] = kp; key[p] = ki; val[i] = vp; val[p] = vi; }
            }
        }
    }
    __syncthreads();
    if (t < NTOP) { cs[blockIdx.x * NTOP + t] = key[t]; ci[blockIdx.x * NTOP + t] = val[t]; }
}

// ---------------------------------------------------------------------------
// Kernel E2: merge 40*100 candidates -> global top 100, emit outputs
// ---------------------------------------------------------------------------
__global__ void topk_stage2(const float* __restrict__ cs, const int* __restrict__ ci,
                            const float* __restrict__ bx, float* __restrict__ out) {
    __shared__ float key[4096];
    __shared__ int   val[4096];
    int t = threadIdx.x;
    for (int p = t; p < 4096; p += blockDim.x) {
        if (p < 40 * NTOP) { key[p] = cs[p]; val[p] = ci[p]; }
        else               { key[p] = NEG_INF; val[p] = 0x7FFFFFFF; }
    }
    for (int k = 2; k <= 4096; k <<= 1) {
        for (int j = k >> 1; j > 0; j >>= 1) {
            __syncthreads();
            for (int q = t; q < 2048; q += blockDim.x) {
                int i = ((q & ~(j - 1)) << 1) | (q & (j - 1));
                int p = i | j;
                bool desc = (i & k) == 0;
                float ki = key[i], kp = key[p];
                int   vi = val[i], vp = val[p];
                bool sw2 = desc ? before_(kp, vp, ki, vi) : before_(ki, vi, kp, vp);
                if (sw2) { key[i] = kp; key[p] = ki; val[i] = vp; val[p] = vi; }
            }
        }
    }
    __syncthreads();
    if (t < NTOP) {
        float s = key[t];
        int pack = val[t];
        int cls = pack >> 11;               // / R (R = 2048)
        int ind = pack & (R - 1);
        out[t] = s;                                   // top_s [100]
        out[NTOP + t * 4 + 0] = bx[ind * 4 + 0];      // top_b [100][4]
        out[NTOP + t * 4 + 1] = bx[ind * 4 + 1];
        out[NTOP + t * 4 + 2] = bx[ind * 4 + 2];
        out[NTOP + t * 4 + 3] = bx[ind * 4 + 3];
        out[5 * NTOP + t] = (float)cls;               // pred_classes [100]
        out[6 * NTOP + t] = (float)ind;               // pred_inds [100]
    }
}

// ---------------------------------------------------------------------------
extern "C" void kernel_launch(void* const* d_in, const int* in_sizes, int n_in,
                              void* d_out, int out_size, void* d_ws, size_t ws_size,
                              hipStream_t stream) {
    const float* boxes  = (const float*)d_in[0];
    const float* scores = (const float*)d_in[1];
    float* out = (float*)d_out;

    char* w = (char*)d_ws;
    const size_t F = sizeof(float);
    float*    bx     = (float*)(w + 0);                       // R*4
    float*    aw     = (float*)(w + 32768);                   // R
    float*    ah     = (float*)(w + 40960);
    float*    aarea  = (float*)(w + 49152);
    unsigned* finite = (unsigned*)(w + 57344);
    size_t cls = (size_t)KCLS * R * F;                        // 163840 B each
    float*    sx1    = (float*)(w + 65536);
    float*    sy1    = (float*)(w + 65536 + 1 * cls);
    float*    sx2    = (float*)(w + 65536 + 2 * cls);
    float*    sy2    = (float*)(w + 65536 + 3 * cls);
    float*    sww    = (float*)(w + 65536 + 4 * cls);
    float*    shh    = (float*)(w + 65536 + 5 * cls);
    float*    sarea  = (float*)(w + 65536 + 6 * cls);
    float*    sscore = (float*)(w + 65536 + 7 * cls);
    unsigned* sidx   = (unsigned*)(w + 65536 + 8 * cls);
    unsigned* svalid = (unsigned*)(w + 65536 + 9 * cls);
    size_t maskOff   = 65536 + 10 * cls;                      // 1703936
    unsigned* mask   = (unsigned*)(w + maskOff);              // KCLS*R*64 u32 = 10.5 MB
    size_t maskBytes = (size_t)KCLS * R * 64 * sizeof(unsigned);
    float*    masked = (float*)(w + maskOff + maskBytes);     // KCLS*R floats
    size_t candOff   = maskOff + maskBytes + (size_t)KCLS * R * F;
    float*    cands  = (float*)(w + candOff);                 // 4000
    int*      candi  = (int*)(w + candOff + 16384);

    hipMemsetAsync(mask, 0, maskBytes, stream);

    nms_preprocess<<<R / 256, 256, 0, stream>>>(boxes, scores, bx, aw, ah, aarea, finite);
    nms_sort_class<<<KCLS, 1024, 0, stream>>>(scores, bx, aw, ah, aarea, finite,
                                              sx1, sy1, sx2, sy2, sww, shh, sarea,
                                              sscore, sidx, svalid);
    nms_iou_mask<<<KCLS * 16, 256, 0, stream>>>(sx1, sy1, sx2, sy2, sww, shh, sarea, mask);
    nms_scan<<<KCLS, 32, 0, stream>>>(mask, svalid, sscore, sidx, masked);
    topk_stage1<<<(KCLS * R) / 1024, 1024, 0, stream>>>(masked, cands, candi);
    topk_stage2<<<1, 1024, 0, stream>>>(cands, candi, bx, out);
}